// RAFT_56538949485363
// MI455X (gfx1250) — compile-verified
//
#include <hip/hip_runtime.h>

// RAFT convex-combination upsampling (SCALE=8) for MI455X / gfx1250.
// N=8, C=48, H=48, W=64  ->  out (8, 48, 384, 512) f32.
//
// Per spatial location: D[48 x 64] = patches[48 x 9] * softmax(mask)[9 x 64].
// Mapped onto V_WMMA_F32_16X16X4_F32 (full f32 precision; kernel is
// store-bandwidth bound at ~363 MB traffic, so fp32 WMMA costs nothing).
// K=9 padded to 12 -> 3 K-chunks; 3 channel tiles x 4 subpixel tiles
// -> 36 WMMAs per location, one wave per location, 8 locations per block.

typedef __attribute__((ext_vector_type(2))) float v2f;
typedef __attribute__((ext_vector_type(8))) float v8f;

#define NN 8
#define CC 48
#define HH 48
#define WW 64
#define SS 8
#define OUT_CSTRIDE (384 * 512) // (SS*HH) * (SS*WW)

__global__ __launch_bounds__(256) void raft_upsample_wmma(
    const float* __restrict__ inp,   // (8, 48, 48, 64)
    const float* __restrict__ mask,  // (8, 576, 48, 64)
    float* __restrict__ out)         // (8, 48, 384, 512)
{
    // LDS staging: mask slab [576 ch][8 w] padded to stride 9 (bank spread),
    // input halo [48 c][3 rows][10 cols] zero-padded.
    __shared__ float smask[576 * 9];
    __shared__ float sinp[48 * 3 * 10];

    const int tid = threadIdx.x;
    const int bid = blockIdx.x;          // 8*48*8 = 3072 blocks
    const int wt  = bid & 7;             // w-tile
    const int h   = (bid >> 3) % HH;
    const int n   = bid / (8 * HH);
    const int w0  = wt * 8;

    // ---- cooperative mask load: 576 ch x 8 w = 4608 = 18*256 ----
    const float* mbase = mask + ((n * 576) * HH + h) * WW + w0;
    #pragma unroll
    for (int i = 0; i < 18; ++i) {
        int idx = tid + i * 256;
        int ch = idx >> 3, wl = idx & 7;
        smask[ch * 9 + wl] = mbase[ch * (HH * WW) + wl];
    }
    // ---- cooperative input halo load: 48*3*10 = 1440 floats ----
    #pragma unroll
    for (int i = 0; i < 6; ++i) {
        int idx = tid + i * 256;
        if (idx < 1440) {
            int c = idx / 30, rem = idx % 30;
            int r = rem / 10, x = rem % 10;
            int hh = h - 1 + r;
            int ww = w0 - 1 + x;
            float v = 0.0f;
            if ((unsigned)hh < (unsigned)HH && (unsigned)ww < (unsigned)WW)
                v = inp[((n * CC + c) * HH + hh) * WW + ww];
            sinp[idx] = v;
        }
    }
    __syncthreads();

    // ---- per-wave GEMM: wave = one w location ----
    const int wl   = tid >> 5;       // wave id 0..7 -> local w
    const int lane = tid & 31;
    const int Lm   = lane & 15;      // M (A) / N (B,D) index within tile
    const int hi   = lane >> 4;      // lane-group: selects K pair / D row half
    const int w    = w0 + wl;

    // B operand (softmaxed mask), ISA B layout for 16x16x4 f32:
    //   lane N = Lm, lanes 0-15 hold K = {v0:Kb, v1:Kb+1} with Kb = 2*hi
    //   per K-chunk base {0,4,8}; K >= 9 is zero padding.
    v2f B[4][3];
    #pragma unroll
    for (int t = 0; t < 4; ++t) {
        const int s = t * 16 + Lm;   // subpixel 0..63
        float mv[9];
        float mx = -1e30f;
        #pragma unroll
        for (int k = 0; k < 9; ++k) {
            mv[k] = smask[(k * 64 + s) * 9 + wl];
            mx = fmaxf(mx, mv[k]);
        }
        float sum = 0.0f;
        #pragma unroll
        for (int k = 0; k < 9; ++k) { mv[k] = __expf(mv[k] - mx); sum += mv[k]; }
        const float inv = __builtin_amdgcn_rcpf(sum);
        #pragma unroll
        for (int k = 0; k < 9; ++k) mv[k] *= inv;

        const float b0a = hi ? mv[2] : mv[0];
        const float b0b = hi ? mv[3] : mv[1];
        const float b1a = hi ? mv[6] : mv[4];
        const float b1b = hi ? mv[7] : mv[5];
        const float b2a = hi ? 0.0f  : mv[8];
        B[t][0] = (v2f){b0a, b0b};
        B[t][1] = (v2f){b1a, b1b};
        B[t][2] = (v2f){b2a, 0.0f};
    }

    const int obase = ((n * CC) * 384 + h * 8) * 512 + w * 8;

    // 3 channel tiles of 16
    #pragma unroll
    for (int ct = 0; ct < 3; ++ct) {
        const float* prow = &sinp[(ct * 16 + Lm) * 30];
        const int kb = 2 * hi;
        // A operand, ISA 16x4 f32 A layout: lane M = Lm,
        // lanes 0-15: {v0:K, v1:K+1} at K=kb; lanes 16-31: K=kb (kb=2).
        // patches(k) = sinp[c][k/3][wl + k%3]; k>=9 -> 0.
        #define PAT(K) (((K) < 9) ? prow[((K) / 3) * 10 + wl + ((K) % 3)] : 0.0f)
        v2f A0 = (v2f){PAT(kb + 0), PAT(kb + 1)};
        v2f A1 = (v2f){PAT(kb + 4), PAT(kb + 5)};
        v2f A2 = (v2f){PAT(kb + 8), PAT(kb + 9)};
        #undef PAT

        v8f acc[4];
        #pragma unroll
        for (int t = 0; t < 4; ++t) acc[t] = (v8f){0,0,0,0,0,0,0,0};

        #pragma unroll
        for (int t = 0; t < 4; ++t) {
            acc[t] = __builtin_amdgcn_wmma_f32_16x16x4_f32(
                false, A0, false, B[t][0], (short)0, acc[t], false, false);
            acc[t] = __builtin_amdgcn_wmma_f32_16x16x4_f32(
                false, A1, false, B[t][1], (short)0, acc[t], false, false);
            acc[t] = __builtin_amdgcn_wmma_f32_16x16x4_f32(
                false, A2, false, B[t][2], (short)0, acc[t], false, false);
        }

        // D layout: VGPR v -> channel row (v + 8*hi), lane -> subpixel s.
        #pragma unroll
        for (int t = 0; t < 4; ++t) {
            const int s = t * 16 + Lm;
            const int p = s >> 3, q = s & 7;
            const int ob = obase + p * 512 + q;
            #pragma unroll
            for (int v = 0; v < 8; ++v) {
                const int cw = ct * 16 + hi * 8 + v;
                out[ob + cw * OUT_CSTRIDE] = acc[t][v];
            }
        }
    }
}

extern "C" void kernel_launch(void* const* d_in, const int* in_sizes, int n_in,
                              void* d_out, int out_size, void* d_ws, size_t ws_size,
                              hipStream_t stream) {
    const float* inp  = (const float*)d_in[0];
    const float* mask = (const float*)d_in[1];
    float* out = (float*)d_out;
    (void)in_sizes; (void)n_in; (void)out_size; (void)d_ws; (void)ws_size;

    dim3 grid(NN * HH * (WW / 8)); // 3072 blocks: (n, h, w-tile)
    dim3 block(256);               // 8 waves, one w location each
    raft_upsample_wmma<<<grid, block, 0, stream>>>(inp, mask, out);
}